// Net_46316927320324
// MI455X (gfx1250) — compile-verified
//
#include <hip/hip_runtime.h>
#include <cstdint>
#include <cstddef>

// ---------------- problem constants (match reference) ----------------
#define N_FEAT 100
#define N_HID  32
#define N_OUT  2
#define KEEP_P 0.8f
#define INV_KEEP 1.25f

typedef __attribute__((ext_vector_type(2))) float v2f;
typedef __attribute__((ext_vector_type(8))) float v8f;

// ---------------- utility: fill float buffer ----------------
__global__ void __launch_bounds__(256) fill_kernel(float* __restrict__ p, float v, long n) {
  long i = (long)blockIdx.x * blockDim.x + threadIdx.x;
  if (i < n) p[i] = v;
}

// ---------------- degree accumulation (self-loop pre-seeded by fill=1.0) ----------------
__global__ void __launch_bounds__(256) deg_kernel(const long long* __restrict__ ei,
                                                  float* __restrict__ deg, long E) {
  long e = (long)blockIdx.x * blockDim.x + threadIdx.x;
  if (e < E) {
    int dst = (int)ei[E + e];
    atomicAdd(&deg[dst], 1.0f);
  }
}

__global__ void __launch_bounds__(256) dinv_kernel(const float* __restrict__ deg,
                                                   float* __restrict__ dinv, long n) {
  long i = (long)blockIdx.x * blockDim.x + threadIdx.x;
  if (i < n) {
    float d = deg[i];
    dinv[i] = (d > 0.0f) ? rsqrtf(d) : 0.0f;
  }
}

// ---------------- GEMM1: h[N,32] = x[N,100] @ W1[100,32] via V_WMMA_F32_16X16X4_F32 ----
// One wave per 16x16 output tile; 2 column tiles cover HIDDEN=32; K loop = 25 steps of 4.
// A (16x4 f32): lanes 0-15 -> rows, VGPR0/1 = K=0/1; lanes 16-31 -> K=2/3.
// B (4x16 f32): lanes 0-15 -> cols with K=0/1; lanes 16-31 -> K=2/3 (mirrors C/D halving).
// C/D (16x16 f32): VGPR r -> row r (lanes 0-15) / row r+8 (lanes 16-31), col = lane&15.
__global__ void __launch_bounds__(128) gemm1_wmma_kernel(const float* __restrict__ x,
                                                         const float* __restrict__ W,
                                                         float* __restrict__ h,
                                                         long n_nodes) {
  const int lane = threadIdx.x & 31;
  const long waveId = (long)blockIdx.x * 4 + (threadIdx.x >> 5);
  const long nRowTiles = (n_nodes + 15) / 16;
  const long rowTile = waveId >> 1;        // 2 column tiles (32/16)
  const int  colTile = (int)(waveId & 1);
  if (rowTile >= nRowTiles) return;        // wave-uniform: EXEC stays all-ones

  const long row0 = rowTile * 16;
  const int  col0 = colTile * 16;
  const long mrow = row0 + (lane & 15);
  const int  koff = (lane >> 4) * 2;       // 0 for lanes 0-15, 2 for lanes 16-31
  const int  ncol = col0 + (lane & 15);

  v8f c = {};
  const float* xr = x + (size_t)mrow * N_FEAT;
#pragma unroll 5
  for (int k0 = 0; k0 < N_FEAT; k0 += 4) {
    v2f a, b;
    a.x = xr[k0 + koff];
    a.y = xr[k0 + koff + 1];
    b.x = W[(size_t)(k0 + koff) * N_HID + ncol];
    b.y = W[(size_t)(k0 + koff + 1) * N_HID + ncol];
    c = __builtin_amdgcn_wmma_f32_16x16x4_f32(false, a, false, b,
                                              (short)0, c, false, false);
  }

  const long mbase = row0 + ((lane >> 4) ? 8 : 0);
#pragma unroll
  for (int r = 0; r < 8; ++r) {
    h[(size_t)(mbase + r) * N_HID + ncol] = c[r];
  }
}

// ---------------- message passing 1: out1[dst,f] += h[src,f]*norm(e) ----------------
// lane-per-feature: 32 consecutive lanes share an edge -> coalesced gather of h[src,:]
// and 32 consecutive float atomics at out1[dst,:]. Stays in L2 (h = 12.8 MB << 192 MB).
__global__ void __launch_bounds__(256) msg1_kernel(const long long* __restrict__ ei,
                                                   const float* __restrict__ dinv,
                                                   const float* __restrict__ h,
                                                   float* __restrict__ out1,
                                                   long E, long N) {
  long g = (long)blockIdx.x * blockDim.x + threadIdx.x;
  long total = (E + N) * N_HID;
  if (g >= total) return;
  long e = g >> 5;              // /32
  int  f = (int)(g & 31);
  int src, dst;
  if (e < E) { src = (int)ei[e]; dst = (int)ei[E + e]; }
  else       { src = dst = (int)(e - E); }   // self loop
  float nrm = dinv[src] * dinv[dst];
  atomicAdd(&out1[(size_t)dst * N_HID + f], h[(size_t)src * N_HID + f] * nrm);
}

// ---------------- Threefry-2x32-20 (JAX-compatible) ----------------
__device__ __forceinline__ unsigned rotl32(unsigned x, int n) {
  return (x << n) | (x >> (32 - n));
}

__device__ __forceinline__ void threefry2x32(unsigned k0, unsigned k1,
                                             unsigned c0, unsigned c1,
                                             unsigned& o0, unsigned& o1) {
  const unsigned ks2 = k0 ^ k1 ^ 0x1BD11BDAu;
  unsigned x0 = c0 + k0, x1 = c1 + k1;
  const int RA[4] = {13, 15, 26, 6};
  const int RB[4] = {17, 29, 16, 24};
#pragma unroll
  for (int g = 0; g < 5; ++g) {
    const bool useB = (g & 1);
#pragma unroll
    for (int r = 0; r < 4; ++r) {
      x0 += x1;
      x1 = rotl32(x1, useB ? RB[r] : RA[r]);
      x1 ^= x0;
    }
    switch (g) {
      case 0: x0 += k1;  x1 += ks2 + 1u; break;
      case 1: x0 += ks2; x1 += k0 + 2u;  break;
      case 2: x0 += k0;  x1 += k1 + 3u;  break;
      case 3: x0 += k1;  x1 += ks2 + 4u; break;
      case 4: x0 += ks2; x1 += k0 + 5u;  break;
    }
  }
  o0 = x0; o1 = x1;
}

// ---------------- bias + ReLU + deterministic dropout (key=(0,42)) ----------------
__global__ void __launch_bounds__(256) relu_drop_kernel(const float* __restrict__ in,
                                                        const float* __restrict__ b1,
                                                        float* __restrict__ h2,
                                                        long N) {
  long i = (long)blockIdx.x * blockDim.x + threadIdx.x;
  long total = N * N_HID;
  if (i >= total) return;
  float v = in[i] + b1[i & 31];
  v = fmaxf(v, 0.0f);

  // JAX threefry_random_bits: counter = iota split in halves
  long half = total >> 1;
  unsigned o0, o1, bits;
  if (i < half) {
    threefry2x32(0u, 42u, (unsigned)i, (unsigned)(i + half), o0, o1);
    bits = o0;
  } else {
    threefry2x32(0u, 42u, (unsigned)(i - half), (unsigned)i, o0, o1);
    bits = o1;
  }
  float u = __uint_as_float(0x3f800000u | (bits >> 9)) - 1.0f;  // [0,1)
  h2[i] = (u < KEEP_P) ? v * INV_KEEP : 0.0f;
}

// ---------------- GEMM2: z[N,2] = h2[N,32] @ W2[32,2] (N=2 too small for WMMA) ------
__global__ void __launch_bounds__(256) gemm2_kernel(const float* __restrict__ h2,
                                                    const float* __restrict__ W2,
                                                    float* __restrict__ z, long N) {
  long n = (long)blockIdx.x * blockDim.x + threadIdx.x;
  if (n >= N) return;
  const float* hr = h2 + (size_t)n * N_HID;
  float a0 = 0.0f, a1 = 0.0f;
#pragma unroll
  for (int k = 0; k < N_HID; ++k) {
    float hv = hr[k];
    a0 = fmaf(hv, W2[k * N_OUT + 0], a0);
    a1 = fmaf(hv, W2[k * N_OUT + 1], a1);
  }
  z[(size_t)n * N_OUT + 0] = a0;
  z[(size_t)n * N_OUT + 1] = a1;
}

// ---------------- message passing 2: out2[dst,:2] += z[src,:2]*norm ----------------
__global__ void __launch_bounds__(256) msg2_kernel(const long long* __restrict__ ei,
                                                   const float* __restrict__ dinv,
                                                   const float* __restrict__ z,
                                                   float* __restrict__ out2,
                                                   long E, long N) {
  long e = (long)blockIdx.x * blockDim.x + threadIdx.x;
  if (e >= E + N) return;
  int src, dst;
  if (e < E) { src = (int)ei[e]; dst = (int)ei[E + e]; }
  else       { src = dst = (int)(e - E); }
  float nrm = dinv[src] * dinv[dst];
  atomicAdd(&out2[(size_t)dst * N_OUT + 0], z[(size_t)src * N_OUT + 0] * nrm);
  atomicAdd(&out2[(size_t)dst * N_OUT + 1], z[(size_t)src * N_OUT + 1] * nrm);
}

// ---------------- bias + log_softmax over 2 classes ----------------
__global__ void __launch_bounds__(256) lsm_kernel(const float* __restrict__ out2,
                                                  const float* __restrict__ b2,
                                                  float* __restrict__ out, long N) {
  long n = (long)blockIdx.x * blockDim.x + threadIdx.x;
  if (n >= N) return;
  float l0 = out2[(size_t)n * N_OUT + 0] + b2[0];
  float l1 = out2[(size_t)n * N_OUT + 1] + b2[1];
  float m = fmaxf(l0, l1);
  float lse = m + logf(expf(l0 - m) + expf(l1 - m));
  out[(size_t)n * N_OUT + 0] = l0 - lse;
  out[(size_t)n * N_OUT + 1] = l1 - lse;
}

// ---------------- host-side orchestration ----------------
static inline unsigned grid_for(long n, int block) {
  return (unsigned)((n + block - 1) / block);
}

extern "C" void kernel_launch(void* const* d_in, const int* in_sizes, int n_in,
                              void* d_out, int out_size, void* d_ws, size_t ws_size,
                              hipStream_t stream) {
  const float*     x   = (const float*)d_in[0];
  const long long* ei  = (const long long*)d_in[1];   // int64 edge_index [2,E]
  const float*     W1  = (const float*)d_in[2];
  const float*     b1  = (const float*)d_in[3];
  const float*     W2  = (const float*)d_in[4];
  const float*     b2  = (const float*)d_in[5];
  float*           out = (float*)d_out;

  const long N = in_sizes[0] / N_FEAT;   // 100000
  const long E = in_sizes[1] / 2;        // 1600000

  // workspace layout (floats)
  float* ws   = (float*)d_ws;
  float* deg  = ws;                       // N
  float* dinv = deg  + N;                 // N
  float* h    = dinv + N;                 // N*32
  float* out1 = h    + (size_t)N * N_HID; // N*32
  float* h2   = out1 + (size_t)N * N_HID; // N*32
  float* z    = h2   + (size_t)N * N_HID; // N*2
  float* out2 = z    + (size_t)N * N_OUT; // N*2

  // 1) seed degree with 1.0 (self loops), zero the accumulators
  fill_kernel<<<grid_for(N, 256), 256, 0, stream>>>(deg, 1.0f, N);
  fill_kernel<<<grid_for((long)N * N_HID, 256), 256, 0, stream>>>(out1, 0.0f, (long)N * N_HID);
  fill_kernel<<<grid_for((long)N * N_OUT, 256), 256, 0, stream>>>(out2, 0.0f, (long)N * N_OUT);

  // 2) degrees + d^{-1/2}
  deg_kernel<<<grid_for(E, 256), 256, 0, stream>>>(ei, deg, E);
  dinv_kernel<<<grid_for(N, 256), 256, 0, stream>>>(deg, dinv, N);

  // 3) h = x @ W1  (WMMA fp32, one wave per 16x16 tile, 4 waves/block)
  {
    long nWaves = ((N + 15) / 16) * 2;    // 2 column tiles for HIDDEN=32
    unsigned blocks = (unsigned)((nWaves + 3) / 4);
    gemm1_wmma_kernel<<<blocks, 128, 0, stream>>>(x, W1, h, N);
  }

  // 4) out1[dst] += h[src]*norm  (edge x feature parallel atomics, L2 resident)
  msg1_kernel<<<grid_for((E + N) * N_HID, 256), 256, 0, stream>>>(ei, dinv, h, out1, E, N);

  // 5) h2 = dropout(relu(out1 + b1))
  relu_drop_kernel<<<grid_for((long)N * N_HID, 256), 256, 0, stream>>>(out1, b1, h2, N);

  // 6) z = h2 @ W2
  gemm2_kernel<<<grid_for(N, 256), 256, 0, stream>>>(h2, W2, z, N);

  // 7) out2[dst] += z[src]*norm
  msg2_kernel<<<grid_for(E + N, 256), 256, 0, stream>>>(ei, dinv, z, out2, E, N);

  // 8) log_softmax(out2 + b2)
  lsm_kernel<<<grid_for(N, 256), 256, 0, stream>>>(out2, b2, out, N);
}